// SimpleCRA_36859409334281
// MI455X (gfx1250) — compile-verified
//
#include <hip/hip_runtime.h>
#include <hip/hip_bf16.h>

// Problem constants (from reference): B=16, L=4096, D=512, WORD_LEN=2
// NW = B*L/2 = 32768 word rows, WORDS = 128 codebook rows, D = 512.
#define DDIM 512
#define NWROWS 32768
#define NWORDS 128

typedef __attribute__((ext_vector_type(2))) float v2f;
typedef __attribute__((ext_vector_type(8))) float v8f;

static __device__ __forceinline__ v8f wmma_f32x4(v2f a, v2f b, v8f c) {
  // D = A(16x4,f32) * B(4x16,f32) + C(16x16,f32)
  return __builtin_amdgcn_wmma_f32_16x16x4_f32(
      /*neg_a=*/false, a, /*neg_b=*/false, b,
      /*c_mod=*/(short)0, c, /*reuse_a=*/false, /*reuse_b=*/false);
}

// ---------------------------------------------------------------------------
// Kernel 0: cnorm[n] = ||codebook[n]||^2   (128 threads, trivial)
// ---------------------------------------------------------------------------
__global__ void __launch_bounds__(128) k_cnorm(const float* __restrict__ CB,
                                               float* __restrict__ cnorm) {
  const int n = threadIdx.x;
  const float* row = CB + (size_t)n * DDIM;
  float s = 0.0f;
  for (int k = 0; k < DDIM; ++k) {
    float c = row[k];
    s += c * c;
  }
  cnorm[n] = s;
}

// ---------------------------------------------------------------------------
// Kernel 1: fused word-mean + distance GEMM (WMMA f32 16x16x4) + argmin +
//           index histogram + codebook gather into output embeddings.
// One wave (32 lanes) owns a 16-row x 128-col score tile.
// Block = 128 threads (4 waves) -> 64 word rows per block; grid = 512.
// ---------------------------------------------------------------------------
__global__ void __launch_bounds__(128) k_dist(
    const float* __restrict__ E,      // char_embeddings [B*L, D]
    const float* __restrict__ CB,     // codebook [128, D]
    const float* __restrict__ cnorm,  // [128]
    int* __restrict__ counts,         // [128] (zeroed)
    float* __restrict__ outIdx,       // [NWROWS] indices as float
    float* __restrict__ outEmb) {     // [NWROWS, D]
  __shared__ float smem[64 * NWORDS];  // 32 KB: per-block score tiles

  const int tid  = threadIdx.x;
  const int lane = tid & 31;
  const int wave = tid >> 5;
  const int half = lane >> 4;   // 0: lanes 0-15, 1: lanes 16-31
  const int l16  = lane & 15;
  const int koff = half * 2;    // fp32 16x4 fragment: upper half holds K+2,K+3

  const int r0   = blockIdx.x * 64 + wave * 16;  // first word row of this wave
  const int rowA = r0 + l16;

  // A fragment source: mean folded out -> A = e(2r) + e(2r+1); score uses
  // cnorm - A.c  ==  ||c||^2 - 2*x.c  (same argmin as squared distance).
  const float* e0 = E + (size_t)(2 * rowA) * DDIM + koff;
  const float* e1 = e0 + DDIM;
  // B fragment: B[k][n] = CB[n][k]; tile t covers columns n = 16t..16t+15.
  const float* bb = CB + (size_t)l16 * DDIM + koff;

  v8f acc[8];
#pragma unroll
  for (int t = 0; t < 8; ++t) acc[t] = (v8f){};

#pragma unroll 2
  for (int k0 = 0; k0 < DDIM; k0 += 4) {
    v2f ea = *(const v2f*)(e0 + k0);
    v2f eb = *(const v2f*)(e1 + k0);
    v2f a  = ea + eb;  // unscaled mean (x2), folded into score
#pragma unroll
    for (int t = 0; t < 8; ++t) {
      v2f b = *(const v2f*)(bb + (size_t)t * (16 * DDIM) + k0);
      acc[t] = wmma_f32x4(a, b, acc[t]);
    }
  }

  // Scatter 16x128 scores into LDS: D-layout VGPR j = row j (lanes<16) / j+8.
#pragma unroll
  for (int t = 0; t < 8; ++t) {
#pragma unroll
    for (int j = 0; j < 8; ++j) {
      int m = wave * 16 + j + half * 8;
      int n = t * 16 + l16;
      smem[m * NWORDS + n] = acc[t][j];
    }
  }
  __syncthreads();

  // Per-row argmin of (cnorm[n] - S[m][n]); lanes 0-15 scan one row each.
  int best = 0;
  if (lane < 16) {
    const float* srow = smem + (wave * 16 + lane) * NWORDS;
    float bv = cnorm[0] - srow[0];
    for (int n = 1; n < NWORDS; ++n) {
      float s = cnorm[n] - srow[n];
      if (s < bv) { bv = s; best = n; }  // strict '<' keeps first index on tie
    }
    outIdx[r0 + lane] = (float)best;
    atomicAdd(&counts[best], 1);
  }

  // Gather selected codebook rows into output embeddings (bulk write, 64 MB).
  for (int j = 0; j < 16; ++j) {
    int w = __shfl(best, j, 32);
    const float4* src = (const float4*)(CB + (size_t)w * DDIM);
    float4* dst = (float4*)(outEmb + (size_t)(r0 + j) * DDIM);
    for (int c = lane; c < DDIM / 4; c += 32) dst[c] = src[c];
  }
}

// ---------------------------------------------------------------------------
// Kernel 2: P = CB @ W^T + b  (128x512x512, WMMA f32), accumulate
//           ssd[m] = sum_n (P[m][n]-CB[m][n])^2 via half-wave shuffle reduce.
// 256 tiles of 16x16 (8 M-tiles x 32 N-tiles); block=128 (4 waves), grid=64.
// ---------------------------------------------------------------------------
__global__ void __launch_bounds__(128) k_proj(const float* __restrict__ CB,
                                              const float* __restrict__ W,
                                              const float* __restrict__ bias,
                                              float* __restrict__ ssd) {
  const int tid  = threadIdx.x;
  const int lane = tid & 31;
  const int wave = tid >> 5;
  const int half = lane >> 4;
  const int l16  = lane & 15;
  const int koff = half * 2;

  const int tile = blockIdx.x * 4 + wave;  // 0..255
  const int mt = tile & 7;                 // M tile
  const int nt = tile >> 3;                // N tile

  // P[m][n] = sum_k CB[m][k] * W[n][k]
  const float* aptr = CB + (size_t)(mt * 16 + l16) * DDIM + koff;
  const float* bptr = W  + (size_t)(nt * 16 + l16) * DDIM + koff;

  v8f acc = (v8f){};
#pragma unroll 4
  for (int k0 = 0; k0 < DDIM; k0 += 4) {
    v2f a = *(const v2f*)(aptr + k0);
    v2f b = *(const v2f*)(bptr + k0);
    acc = wmma_f32x4(a, b, acc);
  }

  const int n = nt * 16 + l16;
  const float bv = bias[n];
#pragma unroll
  for (int j = 0; j < 8; ++j) {
    int m = mt * 16 + j + half * 8;
    float diff = acc[j] + bv - CB[(size_t)m * DDIM + n];
    float v = diff * diff;
    // reduce across each 16-lane half (xor masks < 16 stay within the half)
    v += __shfl_xor(v, 1, 32);
    v += __shfl_xor(v, 2, 32);
    v += __shfl_xor(v, 4, 32);
    v += __shfl_xor(v, 8, 32);
    if (l16 == 0) atomicAdd(&ssd[m], v);
  }
}

// ---------------------------------------------------------------------------
// Kernel 3: loss = sum_w counts[w]*ssd[w] / (NWROWS*D)
// ---------------------------------------------------------------------------
__global__ void __launch_bounds__(128) k_loss(const int* __restrict__ counts,
                                              const float* __restrict__ ssd,
                                              float* __restrict__ outLoss) {
  __shared__ float red[4];
  const int tid = threadIdx.x;
  float v = (float)counts[tid] * ssd[tid];
  v += __shfl_xor(v, 1, 32);
  v += __shfl_xor(v, 2, 32);
  v += __shfl_xor(v, 4, 32);
  v += __shfl_xor(v, 8, 32);
  v += __shfl_xor(v, 16, 32);
  if ((tid & 31) == 0) red[tid >> 5] = v;
  __syncthreads();
  if (tid == 0) {
    float t = red[0] + red[1] + red[2] + red[3];
    outLoss[0] = t * (1.0f / ((float)NWROWS * (float)DDIM));
  }
}

// ---------------------------------------------------------------------------
extern "C" void kernel_launch(void* const* d_in, const int* in_sizes, int n_in,
                              void* d_out, int out_size, void* d_ws,
                              size_t ws_size, hipStream_t stream) {
  // inputs: [0] char_tokens (unused by reference), [1] char_embeddings f32,
  //         [2] word_codebook f32, [3] proj_w f32, [4] proj_b f32
  const float* E  = (const float*)d_in[1];
  const float* CB = (const float*)d_in[2];
  const float* W  = (const float*)d_in[3];
  const float* Bv = (const float*)d_in[4];

  float* out     = (float*)d_out;
  float* outIdx  = out;                                  // 32768 indices
  float* outEmb  = out + NWROWS;                         // 32768*512 floats
  float* outLoss = out + NWROWS + (size_t)NWROWS * DDIM; // scalar

  float* wsf    = (float*)d_ws;
  float* cnorm  = wsf;             // [128]
  float* ssd    = wsf + 128;       // [128]
  int*   counts = (int*)(wsf + 256);  // [128]

  hipMemsetAsync(d_ws, 0, 384 * sizeof(float), stream);  // capturable memset

  k_cnorm<<<1, 128, 0, stream>>>(CB, cnorm);
  k_dist<<<NWROWS / 64, 128, 0, stream>>>(E, CB, cnorm, counts, outIdx, outEmb);
  k_proj<<<64, 128, 0, stream>>>(CB, W, Bv, ssd);
  k_loss<<<1, 128, 0, stream>>>(counts, ssd, outLoss);
}